// AttentionLayer_20074677142123
// MI455X (gfx1250) — compile-verified
//
#include <hip/hip_runtime.h>
#include <hip/hip_bf16.h>

typedef float v2f __attribute__((ext_vector_type(2)));
typedef float v8f __attribute__((ext_vector_type(8)));

#define B_   8
#define T1_  512
#define T2_  128
#define E_   64
#define DL_  256
#define G3_  192   // 3*E

__device__ __forceinline__ float sigmoidf_(float x) { return 1.0f / (1.0f + __expf(-x)); }

// ---------------------------------------------------------------------------
// K0: embedding gather. emb[(b*128+s)*64+e] = emb_table[char_seq[b,s]*64+e]
// ---------------------------------------------------------------------------
__global__ void embed_kernel(const int* __restrict__ char_seq,
                             const float* __restrict__ emb_table,
                             float* __restrict__ emb) {
    int i = blockIdx.x * blockDim.x + threadIdx.x;     // 0 .. 65535
    int row = i >> 6;
    int e   = i & 63;
    int c = char_seq[row];
    emb[i] = emb_table[c * E_ + e];
}

// ---------------------------------------------------------------------------
// K1: xg = emb @ K(dir) + b0(dir)  via V_WMMA_F32_16X16X4_F32.
// grid (64 m-tiles, 12 n-tiles, 2 dirs), block = 32 (one wave).
// A: 16x4 fp32 tile of emb; B: 4x16 fp32 tile of K; chained accumulate over K=64.
// ---------------------------------------------------------------------------
__global__ void xg_wmma_kernel(const float* __restrict__ emb,
                               const float* __restrict__ Kf,
                               const float* __restrict__ Kb,
                               const float* __restrict__ bf,
                               const float* __restrict__ bb,
                               float* __restrict__ xg) {
    int lane = threadIdx.x & 31;
    int ln   = lane & 15;
    int half = lane >> 4;                 // 0: lanes 0-15, 1: lanes 16-31
    int m0   = blockIdx.x << 4;
    int n0   = blockIdx.y << 4;
    int dir  = blockIdx.z;
    const float* Kw = dir ? Kb : Kf;
    const float* bw = dir ? bb : bf;      // row 0 of bias

    int row = m0 + ln;                    // A: M = lane&15 for both halves
    v8f c;
    for (int r = 0; r < 8; ++r) c[r] = 0.0f;

    for (int k0 = 0; k0 < E_; k0 += 4) {
        int ka = k0 + (half << 1);        // K index for VGPR0 (K0 / K2)
        v2f a, bv;
        a.x  = emb[row * E_ + ka];        // K = k0   | k0+2
        a.y  = emb[row * E_ + ka + 1];    // K = k0+1 | k0+3
        bv.x = Kw[ka * G3_ + n0 + ln];        // row K0 | K2, col n
        bv.y = Kw[(ka + 1) * G3_ + n0 + ln];  // row K1 | K3, col n
        c = __builtin_amdgcn_wmma_f32_16x16x4_f32(false, a, false, bv,
                                                  (short)0, c, false, false);
    }
    float bias = bw[n0 + ln];
    for (int r = 0; r < 8; ++r) {
        int mrow = m0 + r + (half << 3);  // D: M = r (lanes0-15) / 8+r (16-31)
        xg[(dir * 1024 + mrow) * G3_ + n0 + ln] = c[r] + bias;
    }
}

// ---------------------------------------------------------------------------
// K2: sequential GRU (Keras reset_after=true, masked).  grid 16 = (b,dir),
// block 192.  R kept in LDS; xg rows streamed from workspace.
// ---------------------------------------------------------------------------
__global__ void gru_kernel(const float* __restrict__ xg,
                           const int* __restrict__ char_seq,
                           const float* __restrict__ Rf,
                           const float* __restrict__ Rb,
                           const float* __restrict__ bf,
                           const float* __restrict__ bb,
                           float* __restrict__ ctx) {
    int blk = blockIdx.x;
    int b   = blk >> 1;
    int dir = blk & 1;
    int j   = threadIdx.x;                // 0..191

    __shared__ float sR[E_ * G3_];        // 48 KB
    __shared__ float sXg[G3_];
    __shared__ float sRg[G3_];
    __shared__ float sH[E_];

    const float* R    = dir ? Rb : Rf;
    const float* bias = dir ? bb : bf;

    for (int i = j; i < E_ * G3_; i += G3_) sR[i] = R[i];
    float b1 = bias[G3_ + j];             // recurrent bias b[1]
    if (j < E_) sH[j] = 0.0f;
    float yprev = 0.0f;
    __syncthreads();

    for (int i = 0; i < T2_; ++i) {
        int s   = dir ? (T2_ - 1 - i) : i;
        int cch = char_seq[b * T2_ + s];
        float xgv = xg[(dir * 1024 + b * T2_ + s) * G3_ + j];
        float rg  = b1;
        for (int k = 0; k < E_; ++k) rg += sH[k] * sR[k * G3_ + j];
        sXg[j] = xgv;
        sRg[j] = rg;
        __syncthreads();
        if (j < E_) {
            float z  = sigmoidf_(sXg[j]        + sRg[j]);
            float r  = sigmoidf_(sXg[E_ + j]   + sRg[E_ + j]);
            float hh = tanhf(sXg[2 * E_ + j] + r * sRg[2 * E_ + j]);
            float hn = z * sH[j] + (1.0f - z) * hh;
            bool  m  = (cch != 0);
            float hout = m ? hn : sH[j];
            float y    = m ? hn : yprev;
            yprev = y;
            sH[j] = hout;
            ctx[(b * T2_ + s) * (2 * E_) + dir * E_ + j] = y;
        }
        __syncthreads();
    }
}

// ---------------------------------------------------------------------------
// K3: per-row dot product + linear index term.
// out[row] = dot(mat[row,:], vec) + coef[0]*(row % mod) + (bias?bias[0]:0)
// one wave per row; block = 256 (8 waves).
// ---------------------------------------------------------------------------
__global__ void row_dot_kernel(const float* __restrict__ mat,
                               const float* __restrict__ vec,
                               float* __restrict__ out,
                               int rows, int cols,
                               const float* __restrict__ coef, int mod,
                               const float* __restrict__ bias) {
    int wave = threadIdx.x >> 5;
    int lane = threadIdx.x & 31;
    int row  = blockIdx.x * (blockDim.x >> 5) + wave;
    if (row >= rows) return;
    float acc = 0.0f;
    for (int c = lane; c < cols; c += 32) acc += mat[row * cols + c] * vec[c];
    for (int off = 16; off > 0; off >>= 1) acc += __shfl_xor(acc, off, 32);
    if (lane == 0) {
        float extra = coef[0] * (float)(row % mod) + (bias ? bias[0] : 0.0f);
        out[row] = acc + extra;
    }
}

// ---------------------------------------------------------------------------
// K4: char_weights output: cw[b,t,s] = l_score[b,t] + c_score[b,s]
// ---------------------------------------------------------------------------
__global__ void charw_kernel(const float* __restrict__ lsc,
                             const float* __restrict__ csc,
                             float* __restrict__ out_cw) {
    int bt = blockIdx.x;                  // 0..4095
    int s  = threadIdx.x;                 // 0..127
    out_cw[bt * T2_ + s] = lsc[bt] + csc[(bt >> 9) * T2_ + s];
}

// ---------------------------------------------------------------------------
// K5: P0 = ctx @ W1_0, P1 = ctx @ W1_1 (bias folded into bc later).
// grid 1024 = (b*128+s), block 64 (o).
// ---------------------------------------------------------------------------
__global__ void p_kernel(const float* __restrict__ ctx,
                         const float* __restrict__ conv1_w,
                         float* __restrict__ P0,
                         float* __restrict__ P1) {
    int row = blockIdx.x;
    int o   = threadIdx.x;
    __shared__ float sctx[2 * E_];
    sctx[o]      = ctx[row * (2 * E_) + o];
    sctx[o + E_] = ctx[row * (2 * E_) + o + E_];
    __syncthreads();
    float a0 = 0.0f, a1 = 0.0f;
    for (int c = 0; c < 2 * E_; ++c) {
        float cv = sctx[c];
        a0 += cv * conv1_w[c * E_ + o];               // W1_0
        a1 += cv * conv1_w[2 * E_ * E_ + c * E_ + o]; // W1_1
    }
    P0[row * E_ + o] = a0;
    P1[row * E_ + o] = a1;
}

// ---------------------------------------------------------------------------
// K6: per-batch banded matrices:
// A[s]=P0[s]@W2_0, B[s]=P1[s]@W2_0 + P0[s+1]@W2_1, C[s]=P1[s+1]@W2_1
// bc[o] = conv1_b @ (W2_0+W2_1) + conv2_b   (block (0,0) writes it)
// grid (126, 8), block 64.
// ---------------------------------------------------------------------------
__global__ void abc_kernel(const float* __restrict__ P0,
                           const float* __restrict__ P1,
                           const float* __restrict__ conv2_w,
                           const float* __restrict__ conv1_b,
                           const float* __restrict__ conv2_b,
                           float* __restrict__ Am,
                           float* __restrict__ Bm,
                           float* __restrict__ Cm,
                           float* __restrict__ bc) {
    int s = blockIdx.x;                   // 0..125
    int b = blockIdx.y;                   // 0..7
    int o = threadIdx.x;                  // 0..63
    __shared__ float p0a[E_], p1a[E_], p0b[E_], p1b[E_];
    int r0 = (b * T2_ + s) * E_;
    int r1 = (b * T2_ + s + 1) * E_;
    p0a[o] = P0[r0 + o];
    p1a[o] = P1[r0 + o];
    p0b[o] = P0[r1 + o];
    p1b[o] = P1[r1 + o];
    __syncthreads();
    float accA = 0.0f, accB = 0.0f, accC = 0.0f;
    for (int c = 0; c < E_; ++c) {
        float w20 = conv2_w[c * E_ + o];
        float w21 = conv2_w[E_ * E_ + c * E_ + o];
        accA += p0a[c] * w20;
        accB += p1a[c] * w20 + p0b[c] * w21;
        accC += p1b[c] * w21;
    }
    int idx = (b * 126 + s) * E_ + o;
    Am[idx] = accA;
    Bm[idx] = accB;
    Cm[idx] = accC;
    if (blockIdx.x == 0 && blockIdx.y == 0) {
        float v = conv2_b[o];
        for (int c = 0; c < E_; ++c)
            v += conv1_b[c] * (conv2_w[c * E_ + o] + conv2_w[E_ * E_ + c * E_ + o]);
        bc[o] = v;
    }
}

// ---------------------------------------------------------------------------
// K7: fused c2 + max-over-s via WMMA.
// c2[t,s,o] = sc[t,s]*A[s,o] + sc[t,s+1]*B[s,o] + sc[t,s+2]*C[s,o]  (K=4, pad 0)
// final[b,t,o] = max_s c2 + bc[o].  grid 256 = (b, t-tile), block 128 = 4 waves
// (one 16x16 o-tile per wave).
// ---------------------------------------------------------------------------
__global__ void final_wmma_kernel(const float* __restrict__ lsc,
                                  const float* __restrict__ csc,
                                  const float* __restrict__ Am,
                                  const float* __restrict__ Bm,
                                  const float* __restrict__ Cm,
                                  const float* __restrict__ bc,
                                  float* __restrict__ out_final) {
    int wave = threadIdx.x >> 5;
    int lane = threadIdx.x & 31;
    int ln   = lane & 15;
    int half = lane >> 4;
    int b    = blockIdx.x >> 5;
    int t0   = (blockIdx.x & 31) << 4;
    int o0   = wave << 4;

    __shared__ float cs[T2_];
    cs[threadIdx.x] = csc[b * T2_ + threadIdx.x];
    __syncthreads();

    float lt = lsc[b * T1_ + t0 + ln];

    v8f acc, cz;
    for (int r = 0; r < 8; ++r) { acc[r] = -__builtin_inff(); cz[r] = 0.0f; }

    for (int s = 0; s < 126; ++s) {
        v2f a, bv;
        // A operand: K0|K2 in .x, K1|K3 in .y
        a.x = lt + cs[s + (half << 1)];                   // sc[t,s] | sc[t,s+2]
        float ay = lt + cs[s + 1];
        a.y = half ? 0.0f : ay;                           // sc[t,s+1] | 0 (K=3)
        int base = (b * 126 + s) * E_ + o0 + ln;
        float b0v = half ? Cm[base] : Am[base];           // row K0 | K2
        float b1v = Bm[base];
        bv.x = b0v;
        bv.y = half ? 0.0f : b1v;                         // row K1 | K3
        v8f d = __builtin_amdgcn_wmma_f32_16x16x4_f32(false, a, false, bv,
                                                      (short)0, cz, false, false);
        for (int r = 0; r < 8; ++r) acc[r] = fmaxf(acc[r], d[r]);
    }

    float bias = bc[o0 + ln];
    for (int r = 0; r < 8; ++r) {
        int trow = t0 + r + (half << 3);
        out_final[(b * T1_ + trow) * E_ + o0 + ln] = acc[r] + bias;
    }
}

// ---------------------------------------------------------------------------
extern "C" void kernel_launch(void* const* d_in, const int* in_sizes, int n_in,
                              void* d_out, int out_size, void* d_ws, size_t ws_size,
                              hipStream_t stream) {
    const float* lstm_outs = (const float*)d_in[0];
    const int*   char_seq  = (const int*)  d_in[1];
    const float* emb_table = (const float*)d_in[2];
    const float* Kf        = (const float*)d_in[3];
    const float* Rf        = (const float*)d_in[4];
    const float* bf        = (const float*)d_in[5];
    const float* Kb        = (const float*)d_in[6];
    const float* Rb        = (const float*)d_in[7];
    const float* bb        = (const float*)d_in[8];
    const float* w_char    = (const float*)d_in[9];
    const float* b_char    = (const float*)d_in[10];
    const float* conv1_w   = (const float*)d_in[11];
    const float* conv1_b   = (const float*)d_in[12];
    const float* conv2_w   = (const float*)d_in[13];
    const float* conv2_b   = (const float*)d_in[14];

    float* out_final = (float*)d_out;                 // (8,512,64)
    float* out_cw    = out_final + B_ * T1_ * E_;     // (8,512,128,1)

    float* ws  = (float*)d_ws;
    float* emb = ws;                 // 65536
    float* xg  = emb + 65536;        // 393216  (2 dirs x 1024 x 192)
    float* ctx = xg + 393216;        // 131072  (8 x 128 x 128)
    float* lsc = ctx + 131072;       // 4096
    float* csc = lsc + 4096;         // 1024
    float* P0  = csc + 1024;         // 65536
    float* P1  = P0 + 65536;         // 65536
    float* Am  = P1 + 65536;         // 64512  (8 x 126 x 64)
    float* Bm  = Am + 64512;         // 64512
    float* Cm  = Bm + 64512;         // 64512
    float* bc  = Cm + 64512;         // 64

    embed_kernel<<<256, 256, 0, stream>>>(char_seq, emb_table, emb);
    xg_wmma_kernel<<<dim3(64, 12, 2), 32, 0, stream>>>(emb, Kf, Kb, bf, bb, xg);
    gru_kernel<<<16, 192, 0, stream>>>(xg, char_seq, Rf, Rb, bf, bb, ctx);
    row_dot_kernel<<<512, 256, 0, stream>>>(lstm_outs, w_char, lsc,
                                            B_ * T1_, DL_, w_char + 384, T1_, b_char);
    row_dot_kernel<<<128, 256, 0, stream>>>(ctx, w_char + 256, csc,
                                            B_ * T2_, 2 * E_, w_char + 385, T2_, nullptr);
    charw_kernel<<<B_ * T1_, T2_, 0, stream>>>(lsc, csc, out_cw);
    p_kernel<<<B_ * T2_, E_, 0, stream>>>(ctx, conv1_w, P0, P1);
    abc_kernel<<<dim3(126, B_), E_, 0, stream>>>(P0, P1, conv2_w, conv1_b, conv2_b,
                                                 Am, Bm, Cm, bc);
    final_wmma_kernel<<<B_ * 32, 128, 0, stream>>>(lsc, csc, Am, Bm, Cm, bc, out_final);
}